// TransformerHybridActor_23106924053158
// MI455X (gfx1250) — compile-verified
//
#include <hip/hip_runtime.h>

typedef __attribute__((ext_vector_type(2))) float v2f;
typedef __attribute__((ext_vector_type(8))) float v8f;

#define B_    1024
#define NT_   128
#define NV_   128
#define NR_   16
#define D_    64
#define H_    32
#define NTGT_ 144   // NR + NV
#define NEG_  (-1e9f)

// ---------------------------------------------------------------------------
// gfx1250 async global->LDS copy (ASYNCcnt-tracked, no VGPR staging).
// VDST = LDS byte address, VADDR = 64-bit global address.
// ---------------------------------------------------------------------------
__device__ __forceinline__ void async_copy_b128(const void* gsrc, void* lds_dst) {
  unsigned loff = (unsigned)(uintptr_t)lds_dst;                 // LDS byte offset
  unsigned long long gaddr = (unsigned long long)(uintptr_t)gsrc;
  asm volatile("global_load_async_to_lds_b128 %0, %1, off"
               :: "v"(loff), "v"(gaddr) : "memory");
}

__device__ __forceinline__ void wait_async_all() {
  asm volatile("s_wait_asynccnt 0x0" ::: "memory");
}

// ---------------------------------------------------------------------------
// Scores kernel: one block (8 waves) per graph.
// scores[b] = (t[b] @ concat(rsu[b], veh[b])^T) * 0.125, masked-fill -1e9.
// ---------------------------------------------------------------------------
__global__ __launch_bounds__(256) void scores_kernel(
    const float* __restrict__ task_emb,   // [B*NT, D]
    const float* __restrict__ veh_emb,    // [B*NV, D]
    const float* __restrict__ rsu_emb,    // [B*NR, D]
    const int*   __restrict__ task_mask,  // [B*NT]
    const int*   __restrict__ target_mask,// [B*NT, NTGT]
    float*       __restrict__ out)        // [B, NT*NTGT]
{
  __shared__ float lA[NT_ * D_];      // 32 KB: task rows
  __shared__ float lT[NTGT_ * D_];    // 36 KB: rsu rows then veh rows

  const int b    = blockIdx.x;
  const int tid  = threadIdx.x;
  const int wave = tid >> 5;          // 0..7  -> M tile
  const int lane = tid & 31;

  // ---- async staging: global -> LDS directly (no VGPR round trip)
  {
    const float4* gA = (const float4*)(task_emb + (size_t)b * NT_ * D_);
    float4* sA = (float4*)lA;
    #pragma unroll
    for (int i = tid; i < NT_ * D_ / 4; i += 256)
      async_copy_b128(gA + i, sA + i);

    const float4* gR = (const float4*)(rsu_emb + (size_t)b * NR_ * D_);
    const float4* gV = (const float4*)(veh_emb + (size_t)b * NV_ * D_);
    float4* sT = (float4*)lT;
    #pragma unroll
    for (int i = tid; i < NR_ * D_ / 4; i += 256)
      async_copy_b128(gR + i, sT + i);
    #pragma unroll
    for (int i = tid; i < NV_ * D_ / 4; i += 256)
      async_copy_b128(gV + i, sT + NR_ * D_ / 4 + i);
  }
  wait_async_all();   // this wave's async copies landed in LDS
  __syncthreads();    // all waves' copies visible

  const int m_base = wave * 16;
  const int half   = lane >> 4;       // 0: K+{0,1}, rows 0..7 ; 1: K+{2,3}, rows 8..15
  const int l15    = lane & 15;

  // ---- preload all 16 A fragments for this wave's M tile (ISA 16x4 f32 layout)
  v2f afrag[16];
  {
    const float* aRow = lA + (m_base + l15) * D_ + 2 * half;
    #pragma unroll
    for (int kk = 0; kk < 16; ++kk)
      afrag[kk] = *(const v2f*)(aRow + kk * 4);
  }

  // ---- hoist task_mask for this wave's 8 rows (contiguous -> b128 loads)
  const size_t rowBase = (size_t)b * NT_ + m_base + 8 * half;   // rows rowBase..rowBase+7
  int tmv[8];
  {
    const int* tp = task_mask + rowBase;
    #pragma unroll
    for (int j = 0; j < 8; ++j) tmv[j] = tp[j];
  }

  // ---- sweep the 9 N tiles
  for (int nt = 0; nt < 9; ++nt) {
    const int n_base = nt * 16;
    const float* bRow = lT + (n_base + l15) * D_ + 2 * half;

    v8f c = {};
    #pragma unroll
    for (int kk = 0; kk < 16; ++kk) {
      v2f bfrag = *(const v2f*)(bRow + kk * 4);
      // D = A(16x4) x B(4x16) + C ; full fp32 accumulation
      c = __builtin_amdgcn_wmma_f32_16x16x4_f32(
          /*neg_a=*/false, afrag[kk],
          /*neg_b=*/false, bfrag,
          /*c_mod=*/(short)0, c,
          /*reuse_a=*/false, /*reuse_b=*/false);
    }

    // ---- fused scale + masked_fill + store
    // C layout: VGPR j -> row m_base + j + 8*half, col n_base + l15.
    // Per-lane base pointer + j*NTGT immediate offsets; branch-free cndmask;
    // NT hints on the single-use mask read and write-once logits.
    const int n = n_base + l15;
    const int*  gm_p  = target_mask + rowBase * NTGT_ + n;
    float*      out_p = out         + rowBase * NTGT_ + n;
    #pragma unroll
    for (int j = 0; j < 8; ++j) {
      const int  gm   = __builtin_nontemporal_load(gm_p + (size_t)j * NTGT_);
      const bool keep = (tmv[j] != 0) & (gm != 0);
      const float val = keep ? c[j] * 0.125f : NEG_;
      __builtin_nontemporal_store(val, out_p + (size_t)j * NTGT_);
    }
  }
}

// ---------------------------------------------------------------------------
// Power head: one wave per graph. lane j = hidden unit j (H == 32 == wave32).
// p = sigmoid( relu([tg ; veh0] @ W1 + b1) @ W2 + b2 )
// ---------------------------------------------------------------------------
__global__ __launch_bounds__(256) void power_kernel(
    const float* __restrict__ task_global, // [B, D]
    const float* __restrict__ veh_emb,     // [B*NV, D]
    const float* __restrict__ W1,          // [2D, H]
    const float* __restrict__ b1,          // [H]
    const float* __restrict__ W2,          // [H, 1]
    const float* __restrict__ b2,          // [1]
    float*       __restrict__ power_out)   // [B]
{
  const int wave = threadIdx.x >> 5;
  const int j    = threadIdx.x & 31;            // hidden unit
  const int b    = blockIdx.x * 8 + wave;
  if (b >= B_) return;

  const float* g  = task_global + (size_t)b * D_;
  const float* v0 = veh_emb + (size_t)b * NV_ * D_;   // first vehicle row

  float acc = b1[j];
  #pragma unroll
  for (int k = 0; k < D_; ++k) acc = fmaf(g[k],  W1[k * H_ + j], acc);
  #pragma unroll
  for (int k = 0; k < D_; ++k) acc = fmaf(v0[k], W1[(D_ + k) * H_ + j], acc);

  float p = fmaxf(acc, 0.0f) * W2[j];
  #pragma unroll
  for (int off = 16; off > 0; off >>= 1) p += __shfl_xor(p, off, 32);

  if (j == 0)
    power_out[b] = 1.0f / (1.0f + __expf(-(p + b2[0])));
}

// ---------------------------------------------------------------------------
extern "C" void kernel_launch(void* const* d_in, const int* in_sizes, int n_in,
                              void* d_out, int out_size, void* d_ws, size_t ws_size,
                              hipStream_t stream) {
  const float* task_emb    = (const float*)d_in[0];
  const float* task_global = (const float*)d_in[1];
  const float* veh_emb     = (const float*)d_in[2];
  const float* rsu_emb     = (const float*)d_in[3];
  const int*   task_mask   = (const int*)d_in[4];
  const int*   target_mask = (const int*)d_in[5];
  const float* W1          = (const float*)d_in[6];
  const float* b1          = (const float*)d_in[7];
  const float* W2          = (const float*)d_in[8];
  const float* b2          = (const float*)d_in[9];

  float* logits = (float*)d_out;                         // [B, NT*NTGT]
  float* power  = logits + (size_t)B_ * NT_ * NTGT_;     // [B]

  scores_kernel<<<B_, 256, 0, stream>>>(task_emb, veh_emb, rsu_emb,
                                        task_mask, target_mask, logits);
  power_kernel<<<B_ / 8, 256, 0, stream>>>(task_global, veh_emb,
                                           W1, b1, W2, b2, power);
}